// ModelNew_11888469475985
// MI455X (gfx1250) — compile-verified
//
#include <hip/hip_runtime.h>
#include <hip/hip_bf16.h>
#include <cstdint>

// ---------------------------------------------------------------------------
// EfficientNet-ish model with the "pool-inside-Sequential" bug:
//   conv1(3->32,3x3,s2)+BN+ReLU -> [64,32,112,112]
//   block1: expand 1x1 (32->96)+BN+ReLU, dw3x3+BN+ReLU, global mean -> [64,96]
//   (everything after is [64,C] dense chains; dw convs see 1x1 -> center tap)
// fp32 everywhere -> V_WMMA_F32_16X16X4_F32 for all GEMM work.
// LDS staging in the fused kernel uses GLOBAL_LOAD_ASYNC_TO_LDS (ASYNCcnt).
// ---------------------------------------------------------------------------

typedef __attribute__((ext_vector_type(2))) float v2f;
typedef __attribute__((ext_vector_type(8))) float v8f;

#define ACT_NONE 0
#define ACT_RELU 1
#define ACT_SIG  2

// ---------------------------------------------------------------------------
// Fold BN: scale = gamma * rsqrt(var+eps); bias = beta - mean*scale.
// If wc != nullptr, additionally scale *= wc[c*9+4] (depthwise center tap,
// valid because the dw input is 1x1 spatial for blocks 2..5).
// ---------------------------------------------------------------------------
__global__ void fold_bn_kernel(const float* __restrict__ g, const float* __restrict__ b,
                               const float* __restrict__ m, const float* __restrict__ v,
                               const float* __restrict__ wc, int C,
                               float* __restrict__ s_out, float* __restrict__ b_out) {
    int i = blockIdx.x * 256 + threadIdx.x;
    if (i >= C) return;
    float s = g[i] * rsqrtf(v[i] + 1e-5f);
    float bb = b[i] - m[i] * s;
    if (wc) s *= wc[i * 9 + 4];
    s_out[i] = s;
    b_out[i] = bb;
}

// ---------------------------------------------------------------------------
// conv1: 3x3 stride-2 pad-1, 3->32 channels, fused folded-BN + ReLU.
// One thread per output pixel computes all 32 channels; weights in LDS.
// Output layout NHWC: X1[((b*112+oh)*112+ow)*32 + co].
// ---------------------------------------------------------------------------
__global__ void conv1_kernel(const float* __restrict__ x, const float* __restrict__ w,
                             const float* __restrict__ sc, const float* __restrict__ bi,
                             float* __restrict__ X1) {
    __shared__ float lw[864];   // (32,3,3,3)
    __shared__ float ls[32], lb[32];
    int tid = threadIdx.x;
    for (int i = tid; i < 864; i += 256) lw[i] = w[i];
    if (tid < 32) { ls[tid] = sc[tid]; lb[tid] = bi[tid]; }
    __syncthreads();

    size_t gid = (size_t)blockIdx.x * 256 + tid;     // 64*112*112 pixels exactly
    int ow = (int)(gid % 112);
    int oh = (int)((gid / 112) % 112);
    int b  = (int)(gid / (112 * 112));

    float acc[32];
#pragma unroll
    for (int co = 0; co < 32; ++co) acc[co] = 0.f;

    for (int ci = 0; ci < 3; ++ci) {
#pragma unroll
        for (int kh = 0; kh < 3; ++kh) {
            int ih = oh * 2 - 1 + kh;
            if (ih < 0 || ih >= 224) continue;
#pragma unroll
            for (int kw = 0; kw < 3; ++kw) {
                int iw = ow * 2 - 1 + kw;
                if (iw < 0 || iw >= 224) continue;
                float v = x[(((size_t)b * 3 + ci) * 224 + ih) * 224 + iw];
                int wi = ci * 9 + kh * 3 + kw;
#pragma unroll
                for (int co = 0; co < 32; ++co)
                    acc[co] += v * lw[co * 27 + wi];
            }
        }
    }
    float* out = X1 + gid * 32;
#pragma unroll
    for (int co = 0; co < 32; ++co)
        out[co] = fmaxf(acc[co] * ls[co] + lb[co], 0.f);
}

// ---------------------------------------------------------------------------
// Fused block-1: expand 1x1 (32->96, WMMA f32) + BN + ReLU into LDS,
// then depthwise 3x3 + BN + ReLU + row pool -> partial sums.
// Tile: one output row h, 28 output cols, needs 3x30 input pixels (halo).
// Halo tile + expand weights are staged into LDS with async-to-LDS loads.
// grid = (4 col-tiles, 112 rows, 64 batch), block = 256 (8 waves).
// ---------------------------------------------------------------------------
__global__ void block1_fused_kernel(const float* __restrict__ X1,
                                    const float* __restrict__ Wexp,   // (96,32)
                                    const float* __restrict__ se, const float* __restrict__ be,
                                    const float* __restrict__ dw_w,   // (96,9)
                                    const float* __restrict__ sd, const float* __restrict__ bd,
                                    float* __restrict__ Ppart) {      // [64][112*4][96]
    // stride 36: keeps 16B alignment for B128 async dests AND spreads banks
    __shared__ __align__(16) float lx[96][36];   // 3x30 halo pixels x 32 ch
    __shared__ float lwt[96][33];                // expand weights, bank-padded
    __shared__ float le[96][97];                 // expand output: pixel x 96 ch

    int tid = threadIdx.x;
    int ct = blockIdx.x;           // 0..3 (28 cols each)
    int h  = blockIdx.y;           // 0..111
    int b  = blockIdx.z;           // batch
    int c0 = ct * 28;

    unsigned long long xbase = (unsigned long long)(uintptr_t)(X1 + (size_t)b * 112 * 112 * 32);
    unsigned long long wbase = (unsigned long long)(uintptr_t)Wexp;

    // ---- stage 1a: async-load halo tile, 16B per lane (4 channels) ----
    for (int i = tid; i < 96 * 8; i += 256) {
        int p   = i >> 3;          // pixel 0..95 (90..95 zero pad)
        int ch4 = (i & 7) * 4;
        unsigned int ldsoff = (unsigned int)(uintptr_t)&lx[p][ch4];
        int r = p / 30, c = p % 30;
        int hh = h - 1 + r, ww = c0 - 1 + c;
        bool valid = (p < 90) && (hh >= 0) && (hh < 112) && (ww >= 0) && (ww < 112);
        if (valid) {
            unsigned int goff = (unsigned int)((((hh * 112 + ww) * 32) + ch4) * 4);
            asm volatile("global_load_async_to_lds_b128 %0, %1, %2"
                         :: "v"(ldsoff), "v"(goff), "s"(xbase) : "memory");
        } else {
            lx[p][ch4 + 0] = 0.f; lx[p][ch4 + 1] = 0.f;
            lx[p][ch4 + 2] = 0.f; lx[p][ch4 + 3] = 0.f;
        }
    }
    // ---- stage 1b: async-load expand weights into LDS (bank-padded) ----
    for (int i = tid; i < 96 * 32; i += 256) {
        int n = i >> 5, k = i & 31;
        unsigned int ldsoff = (unsigned int)(uintptr_t)&lwt[n][k];
        unsigned int goff = (unsigned int)(i * 4);
        asm volatile("global_load_async_to_lds_b32 %0, %1, %2"
                     :: "v"(ldsoff), "v"(goff), "s"(wbase) : "memory");
    }
    asm volatile("s_wait_asynccnt 0x0" ::: "memory");
    __syncthreads();

    // ---- stage 2: expand GEMM [96px x 32] @ [32 x 96ch] via WMMA f32 ----
    int wave = tid >> 5, lane = tid & 31;
    int lm    = lane & 15;
    int khalf = (lane >> 4) * 2;   // 0 or 2
    for (int t = wave; t < 36; t += 8) {          // 6x6 MN tiles over 8 waves
        int mt = t / 6, nt = t % 6;
        int m = mt * 16 + lm;
        int n = nt * 16 + lm;
        v8f acc = {0.f, 0.f, 0.f, 0.f, 0.f, 0.f, 0.f, 0.f};
#pragma unroll
        for (int k = 0; k < 32; k += 4) {
            v2f a, bf;
            a.x  = lx[m][k + khalf];
            a.y  = lx[m][k + khalf + 1];
            bf.x = lwt[n][k + khalf];
            bf.y = lwt[n][k + khalf + 1];
            acc = __builtin_amdgcn_wmma_f32_16x16x4_f32(
                false, a, false, bf, (short)0, acc, false, false);
        }
        float s = se[n], bb = be[n];
#pragma unroll
        for (int r = 0; r < 8; ++r) {
            int mm = mt * 16 + r + ((lane >> 4) << 3);
            le[mm][n] = fmaxf(acc[r] * s + bb, 0.f);
        }
    }
    __syncthreads();

    // ---- stage 2b: zero image-border pixels so dw sees true zero padding.
    // Only rows h-1<0 / h+1>=112 and cols c0-1<0 / c0+28>=112 are invalid.
    int badr = (h == 0) ? 0 : ((h == 111) ? 2 : -1);
    if (badr >= 0)
        for (int i = tid; i < 30 * 96; i += 256)
            le[badr * 30 + (i / 96)][i % 96] = 0.f;
    int badc = (ct == 0) ? 0 : ((ct == 3) ? 29 : -1);
    if (badc >= 0)
        for (int i = tid; i < 3 * 96; i += 256)
            le[(i / 96) * 30 + badc][i % 96] = 0.f;
    __syncthreads();

    // ---- stage 3: depthwise 3x3 + BN + ReLU + row pool (deterministic) ----
    if (tid < 96) {
        int ch = tid;
        float w9[9];
#pragma unroll
        for (int j = 0; j < 9; ++j) w9[j] = dw_w[ch * 9 + j];
        float s = sd[ch], bb = bd[ch];
        float sum = 0.f;
        for (int c = 0; c < 28; ++c) {
            float acc = 0.f;
#pragma unroll
            for (int dr = 0; dr < 3; ++dr)
#pragma unroll
                for (int dc = 0; dc < 3; ++dc)
                    acc += le[dr * 30 + c + dc][ch] * w9[dr * 3 + dc];
            sum += fmaxf(acc * s + bb, 0.f);
        }
        Ppart[(((size_t)b * 112 + h) * 4 + ct) * 96 + ch] = sum;
    }
}

// Fixed-order reduction of the 448 partial row sums per (batch, channel).
__global__ void reduce_pool_kernel(const float* __restrict__ Ppart, float* __restrict__ P) {
    int i = blockIdx.x * 256 + threadIdx.x;    // 64*96 = 6144
    if (i >= 64 * 96) return;
    int ch = i % 96, b = i / 96;
    float s = 0.f;
    for (int j = 0; j < 448; ++j)
        s += Ppart[((size_t)b * 448 + j) * 96 + ch];
    P[i] = s;                                  // mean's /12544 folded into next GEMM
}

// ---------------------------------------------------------------------------
// Generic small GEMM: C[64,N] = act(a_scale*A[64,K] @ W(N,K)^T * s1 + b1),
// optional stage-2 relu(v*s2 + b2) (used to fold the dw center tap + BN).
// One wave per 16x16 tile; K % 4 == 0 for all layers in this network.
// grid = (4, ceil(N/16)), block = 32.
// ---------------------------------------------------------------------------
__global__ void gemm64_kernel(const float* __restrict__ A, const float* __restrict__ W,
                              const float* __restrict__ scale1, const float* __restrict__ bias1,
                              int act1,
                              const float* __restrict__ scale2, const float* __restrict__ bias2,
                              float a_scale, float* __restrict__ C, int N, int K) {
    int lane = threadIdx.x;
    int m0 = blockIdx.x * 16;
    int n0 = blockIdx.y * 16;
    int lm = lane & 15;
    int kh = (lane >> 4) * 2;
    int m = m0 + lm;
    int n = n0 + lm;
    bool nok = (n < N);

    v8f acc = {0.f, 0.f, 0.f, 0.f, 0.f, 0.f, 0.f, 0.f};
    for (int k = 0; k < K; k += 4) {
        v2f a, bf;
        a.x = A[(size_t)m * K + k + kh] * a_scale;
        a.y = A[(size_t)m * K + k + kh + 1] * a_scale;
        bf.x = nok ? W[(size_t)n * K + k + kh] : 0.f;
        bf.y = nok ? W[(size_t)n * K + k + kh + 1] : 0.f;
        acc = __builtin_amdgcn_wmma_f32_16x16x4_f32(
            false, a, false, bf, (short)0, acc, false, false);
    }

    if (nok) {
        float s1 = scale1 ? scale1[n] : 1.f;
        float b1 = bias1 ? bias1[n] : 0.f;
        float s2 = scale2 ? scale2[n] : 1.f;
        float b2 = bias2 ? bias2[n] : 0.f;
#pragma unroll
        for (int r = 0; r < 8; ++r) {
            int mm = m0 + r + ((lane >> 4) << 3);
            float v = acc[r] * s1 + b1;
            if (act1 == ACT_RELU)      v = fmaxf(v, 0.f);
            else if (act1 == ACT_SIG)  v = 1.f / (1.f + __expf(-v));
            if (scale2) v = fmaxf(v * s2 + b2, 0.f);
            C[(size_t)mm * N + n] = v;
        }
    }
}

// ---------------------------------------------------------------------------
// Host-side orchestration
// ---------------------------------------------------------------------------
extern "C" void kernel_launch(void* const* d_in, const int* in_sizes, int n_in,
                              void* d_out, int out_size, void* d_ws, size_t ws_size,
                              hipStream_t stream) {
    static const int CI[5] = {32, 96, 144, 192, 288};
    static const int CO[5] = {96, 144, 192, 288, 384};
    static const int EX[5] = {96, 576, 864, 1152, 1728};

    // ---- enumerate inputs (setup_inputs insertion order, depth-first) ----
    int ii = 0;
    auto IN = [&]() { return (const float*)d_in[ii++]; };
    const float* x  = IN();
    const float* w1 = IN();
    const float *g1 = IN(), *b1 = IN(), *m1 = IN(), *v1 = IN();
    struct BP {
        const float *exp_w, *eg, *eb, *em, *ev;
        const float *dw_w, *dg, *db, *dm, *dv;
        const float *se1, *se2, *proj;
        const float *pg, *pb, *pm, *pv;
    } bp[5];
    for (int i = 0; i < 5; ++i) {
        bp[i].exp_w = IN(); bp[i].eg = IN(); bp[i].eb = IN(); bp[i].em = IN(); bp[i].ev = IN();
        bp[i].dw_w  = IN(); bp[i].dg = IN(); bp[i].db = IN(); bp[i].dm = IN(); bp[i].dv = IN();
        bp[i].se1 = IN(); bp[i].se2 = IN(); bp[i].proj = IN();
        bp[i].pg = IN(); bp[i].pb = IN(); bp[i].pm = IN(); bp[i].pv = IN();
    }
    const float* wf = IN();
    const float *fg = IN(), *fb = IN(), *fm = IN(), *fv = IN();
    const float* fcw = IN();
    const float* fcb = IN();
    (void)in_sizes; (void)n_in; (void)out_size; (void)ws_size;

    // ---- workspace bump allocator (floats) ----
    float* ws = (float*)d_ws;
    size_t off = 0;
    auto alloc = [&](size_t n) { float* p = ws + off; off += n; return p; };
    float* X1    = alloc((size_t)64 * 112 * 112 * 32);   // conv1 out, NHWC (102.8 MB)
    float* Ppart = alloc((size_t)64 * 448 * 96);         // pool partials (11 MB)
    float* P     = alloc(64 * 96);                       // pooled sums
    float* T     = alloc(64 * 1728);                     // y buffer
    float* U     = alloc(64 * 1728);
    float* V     = alloc(64 * 1728);
    float *c1s = alloc(32), *c1b = alloc(32);
    float *es[5], *ebs[5], *ds[5], *dbs[5], *ps[5], *pbs[5];
    for (int i = 0; i < 5; ++i) {
        es[i] = alloc(EX[i]); ebs[i] = alloc(EX[i]);
        ds[i] = alloc(EX[i]); dbs[i] = alloc(EX[i]);
        ps[i] = alloc(CO[i]); pbs[i] = alloc(CO[i]);
    }
    float *fs = alloc(1408), *fbs = alloc(1408);

    // ---- fold BN params on device ----
    auto foldL = [&](const float* g, const float* b, const float* m, const float* v,
                     const float* wc, int C, float* so, float* bo) {
        fold_bn_kernel<<<dim3((C + 255) / 256), dim3(256), 0, stream>>>(g, b, m, v, wc, C, so, bo);
    };
    foldL(g1, b1, m1, v1, nullptr, 32, c1s, c1b);
    for (int i = 0; i < 5; ++i) {
        foldL(bp[i].eg, bp[i].eb, bp[i].em, bp[i].ev, nullptr, EX[i], es[i], ebs[i]);
        // blocks 1..4: fold dw center tap into the dw-BN scale (1x1 spatial input)
        foldL(bp[i].dg, bp[i].db, bp[i].dm, bp[i].dv, (i == 0) ? nullptr : bp[i].dw_w,
              EX[i], ds[i], dbs[i]);
        foldL(bp[i].pg, bp[i].pb, bp[i].pm, bp[i].pv, nullptr, CO[i], ps[i], pbs[i]);
    }
    foldL(fg, fb, fm, fv, nullptr, 1408, fs, fbs);

    // ---- conv1 + BN + ReLU ----
    conv1_kernel<<<dim3(3136), dim3(256), 0, stream>>>(x, w1, c1s, c1b, X1);

    // ---- fused block-1 expand(WMMA)+dw+pool, then reduction ----
    block1_fused_kernel<<<dim3(4, 112, 64), dim3(256), 0, stream>>>(
        X1, bp[0].exp_w, es[0], ebs[0], bp[0].dw_w, ds[0], dbs[0], Ppart);
    reduce_pool_kernel<<<dim3(24), dim3(256), 0, stream>>>(Ppart, P);

    // ---- dense chain on [64, C] matrices (all WMMA f32) ----
    auto gemmL = [&](const float* A, const float* W, const float* s1, const float* bi1,
                     int act, const float* s2, const float* bi2, float ascale,
                     float* Cb, int N, int K) {
        gemm64_kernel<<<dim3(4, (N + 15) / 16), dim3(32), 0, stream>>>(
            A, W, s1, bi1, act, s2, bi2, ascale, Cb, N, K);
    };

    // block 0 SE + proj (pool mean folded in as a_scale = 1/12544)
    gemmL(P, bp[0].se1, nullptr, nullptr, ACT_RELU, nullptr, nullptr,
          1.f / 12544.f, V, 24, 96);
    gemmL(V, bp[0].se2, nullptr, nullptr, ACT_SIG, nullptr, nullptr, 1.f, U, 96, 24);
    gemmL(U, bp[0].proj, ps[0], pbs[0], ACT_NONE, nullptr, nullptr, 1.f, T, 96, 96);

    // blocks 1..4: expand+BNe+ReLU then (dw-center*BNd)+ReLU fused as stage 2
    for (int i = 1; i < 5; ++i) {
        gemmL(T, bp[i].exp_w, es[i], ebs[i], ACT_RELU, ds[i], dbs[i], 1.f, U, EX[i], CI[i]);
        gemmL(U, bp[i].se1, nullptr, nullptr, ACT_RELU, nullptr, nullptr, 1.f, V, EX[i] / 4, EX[i]);
        gemmL(V, bp[i].se2, nullptr, nullptr, ACT_SIG, nullptr, nullptr, 1.f, U, EX[i], EX[i] / 4);
        gemmL(U, bp[i].proj, ps[i], pbs[i], ACT_NONE, nullptr, nullptr, 1.f, T, CO[i], EX[i]);
    }

    // final 1x1 conv + BN + ReLU (384->1408), then FC (1408->1000) + bias
    gemmL(T, wf, fs, fbs, ACT_RELU, nullptr, nullptr, 1.f, U, 1408, 384);
    gemmL(U, fcw, nullptr, fcb, ACT_NONE, nullptr, nullptr, 1.f, (float*)d_out, 1000, 1408);
}